// AmrstrongFrederickViscoplasticity_59897613910133
// MI455X (gfx1250) — compile-verified
//
#include <hip/hip_runtime.h>
#include <stdint.h>

typedef uint32_t u32;
typedef u32 u32x4 __attribute__((ext_vector_type(4)));
typedef u32 u32x8 __attribute__((ext_vector_type(8)));

#define BLOCK 256
#define NEWTON_ITERS 16

// Material constants (fp32)
#define C_SIG0     300.0f
#define C_DSIG     200.0f            // SIGU - SIG0
#define C_BVOCE    10.0f
#define C_KNOR_INV 0.01f             // 1 / K_NORTON
#define C_HKIN     10000.0f
#define C_GAF      50.0f
#define C_TWO_MU   153846.153846f    // 2*MU, MU = E/(2(1+nu))
#define C_LAM      115384.615385f    // E*nu/((1+nu)(1-2nu))
#define C_EPS_EQ   0.002f            // 1e-8 * E

__device__ __forceinline__ float frcp(float x) { return __builtin_amdgcn_rcpf(x); }
__device__ __forceinline__ float fsqrt(float x) { return __builtin_amdgcn_sqrtf(x); }

// ---- Tensor Data Mover descriptors: 1-D f32 tile, 2-group form (<=2D) ----
// Group0: count=1 | lds_addr | global_addr[56:0] | type=2
// Group1: data_size=4B, tile_dim0=n_elem, tensor_dim0 huge (tile always interior)
__device__ __forceinline__ void tdm_desc(u32x4& g0, u32x8& g1,
                                         u32 lds_byte_addr, const void* gptr, u32 n_elem)
{
  uint64_t ga = (uint64_t)(uintptr_t)gptr;
  g0[0] = 1u;                                             // count=1, user descriptor
  g0[1] = lds_byte_addr;                                  // lds_addr
  g0[2] = (u32)ga;                                        // global_addr[31:0]
  g0[3] = (u32)((ga >> 32) & 0x1FFFFFFull) | (2u << 30);  // global_addr[56:32], type=2
  g1[0] = (2u << 16);                // workgroup_mask=0 (not in cluster), data_size=4B
  g1[1] = 0u;                        // atomic barrier off; tensor_dim0[15:0]=0
  g1[2] = 0x4000u | (1u << 16);      // tensor_dim0=0x40000000; tensor_dim1=1
  g1[3] = (n_elem & 0xFFFFu) << 16;  // tensor_dim1 hi=0; tile_dim0=n_elem
  g1[4] = 0u;                        // tile_dim1=0, tile_dim2=0
  g1[5] = 0x40000000u;               // tensor_dim0_stride lo
  g1[6] = 0u;
  g1[7] = 0u;
}

__device__ __forceinline__ void tdm_load_1d(u32 lds_byte_addr, const float* gsrc, u32 n_elem)
{
  u32x4 g0; u32x8 g1;
  tdm_desc(g0, g1, lds_byte_addr, gsrc, n_elem);
  asm volatile("tensor_load_to_lds %0, %1" :: "s"(g0), "s"(g1) : "memory");
}

__device__ __forceinline__ void tdm_store_1d(float* gdst, u32 lds_byte_addr, u32 n_elem)
{
  u32x4 g0; u32x8 g1;
  tdm_desc(g0, g1, lds_byte_addr, gdst, n_elem);
  asm volatile("tensor_store_from_lds %0, %1" :: "s"(g0), "s"(g1) : "memory");
}

__global__ __launch_bounds__(BLOCK, 2)
void afvp_kernel(const float* __restrict__ eps,
                 const float* __restrict__ strain_old,
                 const float* __restrict__ p_old,
                 const float* __restrict__ epsp_old,
                 const float* __restrict__ a1_old,
                 const float* __restrict__ a2_old,
                 const float* __restrict__ stress_old,
                 const float* __restrict__ dt_ptr,
                 float* __restrict__ out, int N)
{
  // Input staging buffers, reused as output staging buffers after the
  // register-load barrier (keeps static LDS at 37 KB).
  __shared__ __align__(16) float s_eps[BLOCK * 6];  // in: eps        out: sig
  __shared__ __align__(16) float s_old[BLOCK * 6];  // in: strain_old out: epsp_new
  __shared__ __align__(16) float s_ep [BLOCK * 6];  // in: epsp_old
  __shared__ __align__(16) float s_a1 [BLOCK * 6];  // in: a1_old     out: a1_new
  __shared__ __align__(16) float s_a2 [BLOCK * 6];  // in: a2_old     out: a2_new
  __shared__ __align__(16) float s_sg [BLOCK * 6];  // in: stress_old
  __shared__ __align__(16) float s_p  [BLOCK];      // in: p_old      out: p_new

  const int tid  = threadIdx.x;
  const int base = blockIdx.x * BLOCK;
  int nrow = N - base; if (nrow > BLOCK) nrow = BLOCK;
  const u32 n6 = (u32)(nrow * 6);

  // ---- Phase 1: TDM stages all inputs into LDS (wave 0 issues; TDM is
  // per-wave, EXEC-independent). Wave 0 drains TENSORcnt, barrier releases all.
  if (tid == 0) {
    tdm_load_1d((u32)(uintptr_t)&s_eps[0], eps        + (size_t)base * 6, n6);
    tdm_load_1d((u32)(uintptr_t)&s_old[0], strain_old + (size_t)base * 6, n6);
    tdm_load_1d((u32)(uintptr_t)&s_ep [0], epsp_old   + (size_t)base * 6, n6);
    tdm_load_1d((u32)(uintptr_t)&s_a1 [0], a1_old     + (size_t)base * 6, n6);
    tdm_load_1d((u32)(uintptr_t)&s_a2 [0], a2_old     + (size_t)base * 6, n6);
    tdm_load_1d((u32)(uintptr_t)&s_sg [0], stress_old + (size_t)base * 6, n6);
    tdm_load_1d((u32)(uintptr_t)&s_p [0],  p_old      + (size_t)base,     (u32)nrow);
    __builtin_amdgcn_s_wait_tensorcnt(0);
  }
  __syncthreads();

  const float dtv = *dt_ptr;       // uniform scalar load
  const int r6 = tid * 6;

  // ---- Phase 2: pull inputs into registers; build trial state & invariants.
  float strial[6], A[6], Bv[6], a1v[6], a2v[6], epv[6];
  float trd = 0.0f;
  float deps[6];
  #pragma unroll
  for (int j = 0; j < 6; ++j) {
    deps[j] = s_eps[r6 + j] - s_old[r6 + j];
    if (j < 3) trd += deps[j];
  }
  #pragma unroll
  for (int j = 0; j < 6; ++j) {
    float s = fmaf(C_TWO_MU, deps[j], s_sg[r6 + j]);
    if (j < 3) s = fmaf(C_LAM, trd, s);
    strial[j] = s;                               // sig_trial = sig_old + C_EL @ deps
  }
  const float trs3 = (strial[0] + strial[1] + strial[2]) * (1.0f / 3.0f);
  #pragma unroll
  for (int j = 0; j < 6; ++j) A[j] = strial[j] - ((j < 3) ? trs3 : 0.0f);  // dev trial
  #pragma unroll
  for (int j = 0; j < 6; ++j) {
    a1v[j] = s_a1[r6 + j]; a2v[j] = s_a2[r6 + j]; epv[j] = s_ep[r6 + j];
  }
  float asum[6];
  #pragma unroll
  for (int j = 0; j < 6; ++j) asum[j] = a1v[j] + a2v[j];
  const float tra3 = (asum[0] + asum[1] + asum[2]) * (1.0f / 3.0f);
  #pragma unroll
  for (int j = 0; j < 6; ++j) Bv[j] = C_HKIN * (asum[j] - ((j < 3) ? tra3 : 0.0f));

  float AA = 0.0f, AB = 0.0f, BB = 0.0f;
  #pragma unroll
  for (int j = 0; j < 6; ++j) {
    AA = fmaf(A[j],  A[j],  AA);
    AB = fmaf(A[j],  Bv[j], AB);
    BB = fmaf(Bv[j], Bv[j], BB);
  }
  AA *= 1.5f; AB *= 1.5f; BB *= 1.5f;            // von-Mises invariants of E(dp)=A-u*B
  const float p0 = s_p[tid];

  __syncthreads();   // all input reads done -> LDS buffers may be overwritten

  // ---- Phase 3: reduced 1-D Newton on dp (register-only, analytic derivative)
  float dp = 0.0f;
  #pragma unroll 4
  for (int it = 0; it < NEWTON_ITERS; ++it) {
    const float u   = frcp(fmaf(C_GAF, dp, 1.0f));          // 1/(1+G*dp)
    const float e2  = fmaf(fmaf(BB, u, -2.0f * AB), u, AA); // |E(dp)|_vm^2
    const float e   = fsqrt(fmaxf(e2, 0.0f));
    const float c   = fmaf(2.0f * C_HKIN, u, C_TWO_MU);     // 2mu + 2H*u
    const float sr  = fmaf(-1.5f * dp, c, e);               // seq (radial return)
    const float seq = fmaxf(sr, C_EPS_EQ);
    const float ex  = __expf(-C_BVOCE * (p0 + dp));
    const float vo  = C_SIG0 + C_DSIG * (1.0f - ex);        // Voce
    const float fy  = fmaxf(seq - vo, 0.0f);
    const float fk  = fy * C_KNOR_INV;
    const float fk2 = fk * fk, fk4 = fk2 * fk2;
    const float r   = fmaf(-dtv, fk4 * fk, dp);             // r = dp - dt*(f/K)^5
    const float du   = -C_GAF * u * u;
    const float inve = (e > 1e-20f) ? frcp(e) : 0.0f;
    const float dedp = fmaf(BB, u, -AB) * du * inve;
    const float dseq = (sr > C_EPS_EQ)
                       ? (dedp - 1.5f * c - 3.0f * C_HKIN * dp * du) : 0.0f;
    const float dvo  = C_DSIG * C_BVOCE * ex;
    const float dnor = (fy > 0.0f) ? 5.0f * fk4 * C_KNOR_INV * (dseq - dvo) : 0.0f;
    const float dr   = fmaf(-dtv, dnor, 1.0f);
    dp = fmaxf(dp - r * frcp(dr), 0.0f);
  }

  // ---- Phase 4: recover tensors, deposit results into LDS staging buffers.
  const float u    = frcp(fmaf(C_GAF, dp, 1.0f));
  const float e2   = fmaf(fmaf(BB, u, -2.0f * AB), u, AA);
  const float e    = fsqrt(fmaxf(e2, 0.0f));
  const float inve = (e > 1e-20f) ? frcp(e) : 0.0f;
  const float kk   = 1.5f * dp * inve;           // n*dp = kk * E_j
  const float dpg  = dp * C_GAF;

  if (tid < nrow) {
    #pragma unroll
    for (int j = 0; j < 6; ++j) {
      const float Ej  = fmaf(-u, Bv[j], A[j]);
      const float ndp = kk * Ej;                            // depsp_j = n_j * dp
      s_eps[r6 + j] = fmaf(-C_TWO_MU, ndp, strial[j]);      // sig
      s_old[r6 + j] = epv[j] + ndp;                         // epsp_new
      s_a1 [r6 + j] = a1v[j] + (ndp - dpg * a1v[j]) * u;    // a1_new
      s_a2 [r6 + j] = a2v[j] + (ndp - dpg * a2v[j]) * u;    // a2_new
    }
    s_p[tid] = p0 + dp;                                     // p_new
  }
  __syncthreads();   // results visible in LDS

  // ---- Phase 5: TDM writes the five contiguous output blocks.
  if (tid == 0) {
    const size_t Ns = (size_t)N;
    tdm_store_1d(out +           (size_t)base * 6, (u32)(uintptr_t)&s_eps[0], n6);
    tdm_store_1d(out +  6 * Ns + (size_t)base,     (u32)(uintptr_t)&s_p [0],  (u32)nrow);
    tdm_store_1d(out +  7 * Ns + (size_t)base * 6, (u32)(uintptr_t)&s_old[0], n6);
    tdm_store_1d(out + 13 * Ns + (size_t)base * 6, (u32)(uintptr_t)&s_a1[0],  n6);
    tdm_store_1d(out + 19 * Ns + (size_t)base * 6, (u32)(uintptr_t)&s_a2[0],  n6);
    __builtin_amdgcn_s_wait_tensorcnt(0);   // LDS stays live until DMA done
  }
}

extern "C" void kernel_launch(void* const* d_in, const int* in_sizes, int n_in,
                              void* d_out, int out_size, void* d_ws, size_t ws_size,
                              hipStream_t stream) {
  const float* eps        = (const float*)d_in[0];
  const float* strain_old = (const float*)d_in[1];
  const float* p_old      = (const float*)d_in[2];
  const float* epsp_old   = (const float*)d_in[3];
  const float* a1_old     = (const float*)d_in[4];
  const float* a2_old     = (const float*)d_in[5];
  const float* stress_old = (const float*)d_in[6];
  const float* dt         = (const float*)d_in[7];
  const int N = in_sizes[2];                     // p_old element count
  dim3 grid((unsigned)((N + BLOCK - 1) / BLOCK)), block(BLOCK);
  hipLaunchKernelGGL(afvp_kernel, grid, block, 0, stream,
                     eps, strain_old, p_old, epsp_old, a1_old, a2_old,
                     stress_old, dt, (float*)d_out, N);
}